// PredictLSTM_42494406427544
// MI455X (gfx1250) — compile-verified
//
#include <hip/hip_runtime.h>
#include <hip/hip_bf16.h>
#include <math.h>

typedef float v2f __attribute__((ext_vector_type(2)));
typedef float v8f __attribute__((ext_vector_type(8)));

#define S_LEN 4096
#define EMB 256
#define HID 128
#define VOCAB 50000
#define HS 132              // padded LDS row stride for h (bank-conflict free)
#define GS 516              // padded LDS row stride for gates

__device__ __forceinline__ v8f wmma4(v2f a, v2f b, v8f c) {
    // D(16x16,f32) = A(16x4,f32) * B(4x16,f32) + C
    return __builtin_amdgcn_wmma_f32_16x16x4_f32(false, a, false, b, (short)0, c, false, false);
}

__device__ __forceinline__ float sigmoidf_(float x) { return 1.0f / (1.0f + expf(-x)); }

// ---------------------------------------------------------------------------
// Kernel 1: suffix sums of token -> end[s] = S - sum_{t>s} tok[t], batches 1..4
// ---------------------------------------------------------------------------
__global__ void suffix_end_kernel(const int* __restrict__ token, int* __restrict__ endbuf) {
    __shared__ int csum[256];
    __shared__ int suf[257];
    int b = blockIdx.x + 1;                  // batches 1..4
    const int* tok = token + (size_t)b * S_LEN;
    int tid = threadIdx.x;
    int lo = tid * 16;
    int s = 0;
    for (int i = 0; i < 16; ++i) s += tok[lo + i];
    csum[tid] = s;
    __syncthreads();
    if (tid == 0) {
        int run = 0;
        for (int i = 255; i >= 0; --i) { suf[i] = run; run += csum[i]; }
    }
    __syncthreads();
    int run = suf[tid];                      // sum of all chunks strictly after this chunk
    int* eb = endbuf + (size_t)blockIdx.x * S_LEN;
    for (int i = 15; i >= 0; --i) {
        int sidx = lo + i;
        eb[sidx] = S_LEN - run;              // end = S - ssum_after
        run += tok[sidx];
    }
}

// ---------------------------------------------------------------------------
// Kernel 2: embedding gather: E0 -> lstm_in[s][0][:], tmp[b'][s][m][:] windows
// ---------------------------------------------------------------------------
__global__ void gather_embed_kernel(const int* __restrict__ x, const int* __restrict__ token,
                                    const float* __restrict__ emb, const int* __restrict__ endbuf,
                                    float* __restrict__ lstm_in, float* __restrict__ tmp) {
    int e = threadIdx.x;                     // 0..255
    int s = blockIdx.x;                      // 0..4095
    int y = blockIdx.y;                      // 0..4
    if (y == 0) {
        int row = x[s];
        lstm_in[((size_t)s * 5 + 0) * EMB + e] = emb[(size_t)row * EMB + e];
        return;
    }
    int b = y;
    int end = endbuf[(size_t)(b - 1) * S_LEN + s];
    int tk  = token[(size_t)b * S_LEN + s];
    int start = end - tk; if (start < 0) start = 0;
    const float* et = emb + (size_t)b * VOCAB * EMB;
    float* dst = tmp + (((size_t)(b - 1) * S_LEN + s) * 4) * EMB + e;
    for (int m = 0; m < 4; ++m) {
        int idx = start + m;
        bool valid = (idx < end) && (end > 0);
        int ci = idx < 0 ? 0 : (idx > S_LEN - 1 ? S_LEN - 1 : idx);
        float val = valid ? et[(size_t)x[ci] * EMB + e] : 0.0f;
        dst[(size_t)m * EMB] = val;
    }
}

// ---------------------------------------------------------------------------
// Kernel 3: conv layer as GEMM via WMMA f32 (dual accumulators).
// act_in: [4][S][Lin][256]; per (b,p): out[m][oc] = tanh(bias + sum_{k=0..511} A[m][k]*W[k][oc])
//   A[m][k] = act_in[b][m][p*256 + k]   (contiguous in k)
//   W[k][oc] = conv2_w[b][l][oc][k&255][k>>8]
// out_mode 0: act_out [4][S][Lout][256];  out_mode 1: lstm_in [s][b+1][256]
// ---------------------------------------------------------------------------
__global__ void conv_gemm_kernel(const float* __restrict__ act_in,
                                 const float* __restrict__ w,
                                 const float* __restrict__ bias,
                                 float* __restrict__ act_out,
                                 int l, int Lin, int Lout, int out_mode) {
    int tid  = threadIdx.x;
    int wave = tid >> 5;
    int lane = tid & 31;
    int half = lane >> 4;
    int lrow = lane & 15;
    int z = blockIdx.z;
    int b = z & 3;
    int p = z >> 2;
    int M0 = (blockIdx.x * 8 + wave) * 16;
    int N0 = blockIdx.y * 16;

    const float* Arow  = act_in + (((size_t)(b * S_LEN + (M0 + lrow)) * Lin + p) * EMB);
    const float* wbase = w + (((size_t)b * 3 + l) * 256 + (N0 + lrow)) * 512;

    v8f acc0 = {};
    v8f acc1 = {};
    for (int k = 0; k < 512; k += 8) {
        int kk0 = k + 2 * half;
        int kk1 = kk0 + 4;
        v2f a0 = *(const v2f*)(Arow + kk0);
        v2f a1 = *(const v2f*)(Arow + kk1);
        int off0 = (kk0 & 255) * 2 + (kk0 >> 8);
        int off1 = (kk1 & 255) * 2 + (kk1 >> 8);
        v2f b0 = { wbase[off0], wbase[off0 + 2] };
        v2f b1 = { wbase[off1], wbase[off1 + 2] };
        acc0 = wmma4(a0, b0, acc0);
        acc1 = wmma4(a1, b1, acc1);
    }
    v8f acc = acc0 + acc1;
    float bv = bias[((size_t)b * 3 + l) * 256 + N0 + lrow];
#pragma unroll
    for (int r = 0; r < 8; ++r) {
        int m = M0 + r + 8 * half;
        float vv = tanhf(acc[r] + bv);
        if (out_mode == 0)
            act_out[(((size_t)(b * S_LEN + m)) * Lout + p) * EMB + N0 + lrow] = vv;
        else
            act_out[((size_t)m * 5 + (b + 1)) * EMB + N0 + lrow] = vv;
    }
}

// ---------------------------------------------------------------------------
// Kernel 4: batched LSTM input projection: G[m][n] = X[m] . wih[n] + bih[n] + bhh[n]
// M = S*5 = 20480, N = 512, K = 256 or 128. X row-major [m][K], wih [n][K].
// ---------------------------------------------------------------------------
__global__ void gemm_gates_kernel(const float* __restrict__ X, int K,
                                  const float* __restrict__ wih,
                                  const float* __restrict__ bih,
                                  const float* __restrict__ bhh,
                                  float* __restrict__ G) {
    int tid  = threadIdx.x;
    int wave = tid >> 5;
    int lane = tid & 31;
    int half = lane >> 4;
    int lrow = lane & 15;
    int M0 = (blockIdx.x * 8 + wave) * 16;
    int N0 = blockIdx.y * 16;

    const float* Arow = X   + (size_t)(M0 + lrow) * K;
    const float* Brow = wih + (size_t)(N0 + lrow) * K;

    v8f acc0 = {};
    v8f acc1 = {};
    for (int k = 0; k < K; k += 8) {
        int kk0 = k + 2 * half;
        int kk1 = kk0 + 4;
        v2f a0 = *(const v2f*)(Arow + kk0);
        v2f a1 = *(const v2f*)(Arow + kk1);
        v2f b0 = *(const v2f*)(Brow + kk0);
        v2f b1 = *(const v2f*)(Brow + kk1);
        acc0 = wmma4(a0, b0, acc0);
        acc1 = wmma4(a1, b1, acc1);
    }
    v8f acc = acc0 + acc1;
    float bs = bih[N0 + lrow] + bhh[N0 + lrow];
#pragma unroll
    for (int r = 0; r < 8; ++r)
        G[(size_t)(M0 + r + 8 * half) * 512 + N0 + lrow] = acc[r] + bs;
}

// ---------------------------------------------------------------------------
// Kernel 5: sequential LSTM recurrence. One 1024-thread workgroup (32 waves).
// whh^T (128x512 f32, 256KB) resident in LDS; h padded to 16 rows (stride 132).
// Per step: wave w computes gate columns [16w,16w+16) via 2 interleaved chains
// of wmma_f32_16x16x4, gates staged in LDS, 640 threads do the cell update.
// Next step's precomputed gates are prefetched (global_prefetch) during update.
// ---------------------------------------------------------------------------
__global__ void lstm_recur_kernel(const float* __restrict__ G,     // [S][5][512]
                                  const float* __restrict__ whh,   // [512][128]
                                  float* __restrict__ ys,          // [S][5][128] or nullptr
                                  float* __restrict__ hfin,        // [640]
                                  float* __restrict__ cfin) {      // [640]
    extern __shared__ float sm[];
    float* whh_s = sm;                        // 128*512 = 65536 : whh_s[k*512+n] = whh[n][k]
    float* h_s   = sm + 65536;                // 16*HS = 2112
    float* c_s   = h_s + 16 * HS;             // 640
    float* g_s   = c_s + 640;                 // 5*GS = 2580

    int tid = threadIdx.x;
    for (int i = tid; i < 65536; i += 1024) {
        int n = i >> 7, k = i & 127;
        whh_s[k * 512 + n] = whh[i];
    }
    for (int i = tid; i < 16 * HS; i += 1024) h_s[i] = 0.0f;
    for (int i = tid; i < 640; i += 1024)     c_s[i] = 0.0f;
    __syncthreads();

    int wave = tid >> 5;
    int lane = tid & 31;
    int half = lane >> 4;
    int lrow = lane & 15;
    int N0 = wave * 16;

    for (int t = 0; t < S_LEN; ++t) {
        v8f acc0 = {};
        v8f acc1 = {};
#pragma unroll
        for (int k = 0; k < HID; k += 8) {
            int kk0 = k + 2 * half;
            int kk1 = kk0 + 4;
            v2f a0 = { h_s[lrow * HS + kk0], h_s[lrow * HS + kk0 + 1] };
            v2f a1 = { h_s[lrow * HS + kk1], h_s[lrow * HS + kk1 + 1] };
            v2f b0 = { whh_s[(size_t)kk0 * 512 + N0 + lrow],
                       whh_s[(size_t)(kk0 + 1) * 512 + N0 + lrow] };
            v2f b1 = { whh_s[(size_t)kk1 * 512 + N0 + lrow],
                       whh_s[(size_t)(kk1 + 1) * 512 + N0 + lrow] };
            acc0 = wmma4(a0, b0, acc0);
            acc1 = wmma4(a1, b1, acc1);
        }
        v8f acc = acc0 + acc1;
        if (half == 0) {
            const float* Gt = G + ((size_t)t * 5) * 512 + N0 + lrow;
#pragma unroll
            for (int r = 0; r < 5; ++r)
                g_s[r * GS + N0 + lrow] = acc[r] + Gt[(size_t)r * 512];
        }
        __syncthreads();
        if (tid < 640) {
            int m = tid >> 7, j = tid & 127;
            // hide next step's gate fetch behind the barriers
            if (t + 1 < S_LEN)
                __builtin_prefetch(G + ((size_t)(t + 1) * 5) * 512 + (size_t)tid * 4, 0, 0);
            float gi = g_s[m * GS + j];
            float gf = g_s[m * GS + j + 128];
            float gg = g_s[m * GS + j + 256];
            float go = g_s[m * GS + j + 384];
            float c = sigmoidf_(gf) * c_s[tid] + sigmoidf_(gi) * tanhf(gg);
            c_s[tid] = c;
            float h = sigmoidf_(go) * tanhf(c);
            h_s[m * HS + j] = h;
            if (ys) ys[((size_t)t * 5 + m) * HID + j] = h;
        }
        __syncthreads();
    }
    if (tid < 640) {
        int m = tid >> 7, j = tid & 127;
        hfin[tid] = h_s[m * HS + j];
        cfin[tid] = c_s[tid];
    }
}

// ---------------------------------------------------------------------------
// Kernel 6: copy h/c finals to output + tiny strided conv chain on cell states
// ---------------------------------------------------------------------------
__global__ void finalize_kernel(const float* __restrict__ hf,   // [2][5][128]
                                const float* __restrict__ cf,   // [2][5][128]
                                const float* __restrict__ pw0,  // [1][2][2]
                                const float* __restrict__ pb0,  // [1]
                                const float* __restrict__ pwr,  // [6][1][1][2]
                                const float* __restrict__ pbr,  // [6][1]
                                float* __restrict__ out) {
    int tid = threadIdx.x;
    for (int i = tid; i < 1280; i += 256) out[5 + i] = hf[i];
    for (int i = tid; i < 1280; i += 256) out[5 + 1280 + i] = cf[i];
    if (tid < 5) {
        int b = tid;
        float v[64];
        for (int p = 0; p < 64; ++p) {
            float s = pb0[0];
            for (int l = 0; l < 2; ++l)
                for (int k = 0; k < 2; ++k)
                    s += cf[(size_t)l * 640 + b * 128 + 2 * p + k] * pw0[l * 2 + k];
            v[p] = tanhf(s);
        }
        int len = 64;
        for (int st = 0; st < 6; ++st) {
            len >>= 1;
            for (int p = 0; p < len; ++p)
                v[p] = tanhf(pbr[st] + v[2 * p] * pwr[st * 2] + v[2 * p + 1] * pwr[st * 2 + 1]);
        }
        out[b] = v[0];
    }
}

// ---------------------------------------------------------------------------
extern "C" void kernel_launch(void* const* d_in, const int* in_sizes, int n_in,
                              void* d_out, int out_size, void* d_ws, size_t ws_size,
                              hipStream_t stream) {
    (void)in_sizes; (void)n_in; (void)out_size; (void)ws_size;
    const int*   x      = (const int*)d_in[0];
    const int*   token  = (const int*)d_in[1];
    // d_in[2..4] = h_state, c_state, pos : ignored by the reference math
    const float* emb    = (const float*)d_in[5];
    const float* conv2w = (const float*)d_in[6];
    const float* conv2b = (const float*)d_in[7];
    const float* wih0   = (const float*)d_in[8];
    const float* whh0   = (const float*)d_in[9];
    const float* bih0   = (const float*)d_in[10];
    const float* bhh0   = (const float*)d_in[11];
    const float* wih1   = (const float*)d_in[12];
    const float* whh1   = (const float*)d_in[13];
    const float* bih1   = (const float*)d_in[14];
    const float* bhh1   = (const float*)d_in[15];
    const float* pw0    = (const float*)d_in[16];
    const float* pb0    = (const float*)d_in[17];
    const float* pwr    = (const float*)d_in[18];
    const float* pbr    = (const float*)d_in[19];
    float* out = (float*)d_out;
    float* ws  = (float*)d_ws;

    // workspace layout (float offsets), with region reuse:
    const size_t OFF_LSTMIN = 0;          // 5*4096*256        = 5,242,880
    const size_t OFF_TMP    = 5242880;    // 4*4096*4*256      = 16,777,216 (reused: a2, then G)
    const size_t OFF_A1     = 22020096;   // 4*4096*3*256      = 12,582,912 (reused: ys)
    const size_t OFF_END    = 34603008;   // 4*4096 ints
    const size_t OFF_HFIN   = 34619392;   // 2*640
    const size_t OFF_CFIN   = 34620672;   // 2*640

    float* lstm_in = ws + OFF_LSTMIN;
    float* tmp     = ws + OFF_TMP;
    float* a1      = ws + OFF_A1;
    float* a2      = ws + OFF_TMP;   // tmp dead after layer-0 conv
    float* G       = ws + OFF_TMP;   // a2 dead after layer-2 conv
    float* ys      = ws + OFF_A1;    // a1 dead after layer-1 conv
    int*   endb    = (int*)(ws + OFF_END);
    float* hfin    = ws + OFF_HFIN;
    float* cfin    = ws + OFF_CFIN;

    const size_t LDS_BYTES = (size_t)(65536 + 16 * HS + 640 + 5 * GS) * sizeof(float); // ~277 KB

    suffix_end_kernel<<<4, 256, 0, stream>>>(token, endb);
    gather_embed_kernel<<<dim3(S_LEN, 5, 1), 256, 0, stream>>>(x, token, emb, endb, lstm_in, tmp);

    // conv stack: 3 layers, each a set of (M=4096, K=512, N=256) WMMA GEMMs per (b,p)
    conv_gemm_kernel<<<dim3(32, 16, 12), 256, 0, stream>>>(tmp, conv2w, conv2b, a1, 0, 4, 3, 0);
    conv_gemm_kernel<<<dim3(32, 16,  8), 256, 0, stream>>>(a1,  conv2w, conv2b, a2, 1, 3, 2, 0);
    conv_gemm_kernel<<<dim3(32, 16,  4), 256, 0, stream>>>(a2,  conv2w, conv2b, lstm_in, 2, 2, 1, 1);

    // LSTM layer 0: batched input GEMM + sequential recurrence (weights in LDS)
    gemm_gates_kernel<<<dim3(160, 32, 1), 256, 0, stream>>>(lstm_in, 256, wih0, bih0, bhh0, G);
    lstm_recur_kernel<<<1, 1024, LDS_BYTES, stream>>>(G, whh0, ys, hfin, cfin);

    // LSTM layer 1
    gemm_gates_kernel<<<dim3(160, 32, 1), 256, 0, stream>>>(ys, 128, wih1, bih1, bhh1, G);
    lstm_recur_kernel<<<1, 1024, LDS_BYTES, stream>>>(G, whh1, nullptr, hfin + 640, cfin + 640);

    finalize_kernel<<<1, 256, 0, stream>>>(hfin, cfin, pw0, pb0, pwr, pbr, out);
}